// Model_15307263443698
// MI455X (gfx1250) — compile-verified
//
#include <hip/hip_runtime.h>
#include <cstdint>

#define BLOCK 256
#define MAX_NODES 1024
#define STACK_DEPTH 16

typedef __attribute__((ext_vector_type(4))) unsigned int u32x4;
typedef __attribute__((ext_vector_type(8))) int          i32x8;
typedef __attribute__((ext_vector_type(4))) int          i32x4;

#if __has_builtin(__builtin_amdgcn_tensor_load_to_lds)
#define HAVE_TDM 1
#else
#define HAVE_TDM 0
#endif

#if HAVE_TDM
// Issue one TDM DMA: contiguous 1-D array of `ndwords` dwords, global -> LDS.
// Descriptor bit layout per CDNA5 ISA §8 (async_tensor doc).
__device__ __forceinline__ void tdm_load_1d(uint32_t lds_byte_off,
                                            const void* gptr,
                                            uint32_t ndwords) {
  uint64_t ga = (uint64_t)(uintptr_t)gptr;
  u32x4 g0;
  g0.x = 1u;                                   // count=1 (valid descriptor)
  g0.y = lds_byte_off;                         // lds_addr [63:32]
  g0.z = (uint32_t)ga;                         // global_addr [95:64]
  g0.w = ((uint32_t)(ga >> 32) & 0x01FFFFFFu)  // global_addr [120:96]
         | (2u << 30);                         // type=2 ("image") [127:126]
  i32x8 g1;
  g1[0] = 0x00020000;                          // wg_mask=0, data_size=2 (4B)
  g1[1] = (int)((ndwords & 0xFFFFu) << 16);    // tensor_dim0[15:0]
  g1[2] = (int)((ndwords >> 16) | (1u << 16)); // tensor_dim0[31:16], tensor_dim1=1
  g1[3] = (int)((ndwords & 0xFFFFu) << 16);    // tile_dim0 = ndwords (<65536)
  g1[4] = 1;                                   // tile_dim1=1, tile_dim2=0
  g1[5] = (int)ndwords;                        // tensor_dim0_stride[31:0]
  g1[6] = 0;                                   // stride0 hi, stride1 lo
  g1[7] = 0;
  i32x4 z4 = {0, 0, 0, 0};
#if defined(__clang_major__) && (__clang_major__ >= 23)
  i32x8 z8 = {0, 0, 0, 0, 0, 0, 0, 0};
  __builtin_amdgcn_tensor_load_to_lds(g0, g1, z4, z4, z8, 0);
#else
  __builtin_amdgcn_tensor_load_to_lds(g0, g1, z4, z4, 0);
#endif
}
#endif

__global__ __launch_bounds__(BLOCK) void bvh_traverse_kernel(
    const float* __restrict__ ro, const float* __restrict__ rd,
    const float* __restrict__ bmin, const float* __restrict__ bmax,
    const int* __restrict__ left, const int* __restrict__ right,
    float* __restrict__ out, int n_rays, int num_nodes) {
  __shared__ float s_min[3 * MAX_NODES];
  __shared__ float s_max[3 * MAX_NODES];
  __shared__ int   s_left[MAX_NODES];
  __shared__ int   s_right[MAX_NODES];
  __shared__ int   s_stack[STACK_DEPTH * BLOCK];

  const int tid = threadIdx.x;
  const uint32_t nd3 = (uint32_t)(3 * num_nodes);
  const uint32_t nd1 = (uint32_t)num_nodes;

#if HAVE_TDM
  if (tid == 0) {
    tdm_load_1d((uint32_t)(uintptr_t)(const void*)s_min,   bmin,  nd3);
    tdm_load_1d((uint32_t)(uintptr_t)(const void*)s_max,   bmax,  nd3);
    tdm_load_1d((uint32_t)(uintptr_t)(const void*)s_left,  left,  nd1);
    tdm_load_1d((uint32_t)(uintptr_t)(const void*)s_right, right, nd1);
    __builtin_amdgcn_s_wait_tensorcnt(0);   // TENSORcnt -> 0: DMA landed in LDS
  }
#else
  for (uint32_t i = tid; i < nd3; i += BLOCK) { s_min[i] = bmin[i]; s_max[i] = bmax[i]; }
  for (uint32_t i = tid; i < nd1; i += BLOCK) { s_left[i] = left[i]; s_right[i] = right[i]; }
#endif
  __syncthreads();

  const int gstride = gridDim.x * BLOCK;
  for (int ray = blockIdx.x * BLOCK + tid; ray < n_rays; ray += gstride) {
    // Prefetch next grid-stride ray (gfx1250 global_prefetch_b8).
    if (ray + gstride < n_rays) {
      __builtin_prefetch(&ro[3 * (ray + gstride)], 0, 0);
      __builtin_prefetch(&rd[3 * (ray + gstride)], 0, 0);
    }
    const float ox = ro[3 * ray + 0], oy = ro[3 * ray + 1], oz = ro[3 * ray + 2];
    const float dx = rd[3 * ray + 0], dy = rd[3 * ray + 1], dz = rd[3 * ray + 2];
    const float ix = 1.0f / (dx + 1e-10f);
    const float iy = 1.0f / (dy + 1e-10f);
    const float iz = 1.0f / (dz + 1e-10f);

    float closest = __builtin_inff();
    int sp = 1;
    s_stack[tid] = 0;  // push root

    while (sp > 0) {
      sp--;
      const int node = s_stack[sp * BLOCK + tid];
      const float bx0 = s_min[3 * node + 0], by0 = s_min[3 * node + 1], bz0 = s_min[3 * node + 2];
      const float bx1 = s_max[3 * node + 0], by1 = s_max[3 * node + 1], bz1 = s_max[3 * node + 2];
      const float t0x = (bx0 - ox) * ix, t1x = (bx1 - ox) * ix;
      const float t0y = (by0 - oy) * iy, t1y = (by1 - oy) * iy;
      const float t0z = (bz0 - oz) * iz, t1z = (bz1 - oz) * iz;
      const float tn = fmaxf(fmaxf(fminf(t0x, t1x), fminf(t0y, t1y)), fminf(t0z, t1z));
      const float tf = fminf(fminf(fmaxf(t0x, t1x), fmaxf(t0y, t1y)), fmaxf(t0z, t1z));
      if ((tn <= tf) && (tf >= 0.0f) && (tn < closest)) {
        const int l = s_left[node];
        if (l < 0) {
          closest = fminf(closest, fmaxf(tn, 0.0f));   // leaf hit
        } else {
          s_stack[sp * BLOCK + tid] = l; sp++;         // push left
          const int r = s_right[node];
          if (r >= 0) { s_stack[sp * BLOCK + tid] = r; sp++; }  // push right
        }
      }
    }
    out[ray] = closest;
  }
}

extern "C" void kernel_launch(void* const* d_in, const int* in_sizes, int n_in,
                              void* d_out, int out_size, void* d_ws, size_t ws_size,
                              hipStream_t stream) {
  const float* ro   = (const float*)d_in[0];
  const float* rd   = (const float*)d_in[1];
  const float* bmin = (const float*)d_in[2];
  const float* bmax = (const float*)d_in[3];
  const int*   left = (const int*)d_in[4];
  const int*   rght = (const int*)d_in[5];
  // d_in[6] (is_leaf) is redundant: leaf <=> left < 0.

  const int n_rays    = in_sizes[0] / 3;
  const int num_nodes = in_sizes[2] / 3;  // 1023, fits MAX_NODES

  const int grid = (n_rays + BLOCK - 1) / BLOCK;
  bvh_traverse_kernel<<<grid, BLOCK, 0, stream>>>(
      ro, rd, bmin, bmax, left, rght, (float*)d_out, n_rays, num_nodes);
}